// NeuralRenderer_37271726195154
// MI455X (gfx1250) — compile-verified
//
#include <hip/hip_runtime.h>
#include <stdint.h>

// Problem constants (fixed by the reference)
#define SIGMA_   1e-4f
#define GAMMA_   1e-4f
#define ZNEAR_   1.0f
#define ZFAR_    100.0f
#define EPSV_    1e-10f

#define TPB   256
#define KF    8
#define DF    16
#define HW_   65536          // 256*256
#define NCH   17             // D + alpha
#define NPIX  262144         // 4*256*256

// Low 32 bits of a generic pointer to an LDS object are the LDS byte offset
// (ISA 10.2: LDS aperture address is truncated to addr[31:0]).
__device__ __forceinline__ uint32_t lds_off_of(const void* p) {
  return (uint32_t)(uintptr_t)p;
}

// gfx1250 async DMA: 16B global -> LDS, tracked with ASYNCcnt.
__device__ __forceinline__ void async_copy_b128(uint32_t lds_byte, const void* gsrc) {
  asm volatile("global_load_async_to_lds_b128 %0, %1, off"
               :: "v"(lds_byte), "v"((unsigned long long)(uintptr_t)gsrc)
               : "memory");
}

__device__ __forceinline__ void wait_async0() {
#if __has_builtin(__builtin_amdgcn_s_wait_asynccnt)
  __builtin_amdgcn_s_wait_asynccnt(0);
#else
  asm volatile("s_wait_asynccnt 0" ::: "memory");
#endif
}

__global__ __launch_bounds__(TPB, 4)
void softblend_kernel(const float* __restrict__ vert,    // (V,16)
                      const float* __restrict__ bary,    // (NPIX,8,3)
                      const float* __restrict__ dists,   // (NPIX,8)
                      const float* __restrict__ zbuf,    // (NPIX,8)
                      const int*   __restrict__ faces,   // (F,3)
                      const int*   __restrict__ ptfg,    // (NPIX,8)
                      float*       __restrict__ out)     // (N,17,HW)
{
  __shared__ int   s_ptf[TPB * KF];   // 8 KB
  __shared__ float s_dst[TPB * KF];   // 8 KB
  __shared__ float s_zbf[TPB * KF];   // 8 KB

  const int t  = threadIdx.x;
  const int p0 = blockIdx.x * TPB;    // first pixel of this block
  const int p  = p0 + t;              // this thread's pixel

  // ---- stage pass-1 streams into LDS with async DMA (ASYNCcnt path) ----
  {
    const char* gp = (const char*)(ptfg  + (size_t)p0 * KF);
    const char* gd = (const char*)(dists + (size_t)p0 * KF);
    const char* gz = (const char*)(zbuf  + (size_t)p0 * KF);
    const uint32_t lp = lds_off_of(s_ptf);
    const uint32_t ld = lds_off_of(s_dst);
    const uint32_t lz = lds_off_of(s_zbf);
    const uint32_t o0 = 16u * (uint32_t)t;          // each region: 8KB = 512 x 16B
    const uint32_t o1 = o0 + 16u * TPB;             // 2 chunks per thread
    async_copy_b128(lp + o0, gp + o0);
    async_copy_b128(lp + o1, gp + o1);
    async_copy_b128(ld + o0, gd + o0);
    async_copy_b128(ld + o1, gd + o1);
    async_copy_b128(lz + o0, gz + o0);
    async_copy_b128(lz + o1, gz + o1);
  }

  // Warm the bary stream for pass 2 while the DMA is in flight.
  const float* baryP = bary + (size_t)p * (KF * 3);   // 96 B / thread
  __builtin_prefetch(baryP, 0, 3);
  __builtin_prefetch(baryP + 16, 0, 3);

  wait_async0();        // this wave's DMAs complete
  __syncthreads();      // all waves' DMAs visible

  // ---------------- pass 1: prob / z_inv / max ----------------
  int   fidx[KF];
  float prob[KF];
  float zinv[KF];
  float omp  = 1.0f;    // prod(1 - prob)
  float zmax = 0.0f;

  const float inv_sigma = 1.0f / (SIGMA_ + 1e-8f);
  const float inv_span  = 1.0f / (ZFAR_ - ZNEAR_);
  const float inv_gamma = 1.0f / GAMMA_;

#pragma unroll
  for (int k = 0; k < KF; ++k) {
    const int   f = s_ptf[t * KF + k];
    const float d = s_dst[t * KF + k];
    const float z = s_zbf[t * KF + k];
    const bool  m = (f >= 0);
    // sigmoid(-d/sigma) = 1 / (1 + exp(d/sigma)), masked
    const float pr = m ? (1.0f / (1.0f + __expf(d * inv_sigma))) : 0.0f;
    prob[k] = pr;
    omp *= (1.0f - pr);
    const float zi = m ? (ZFAR_ - z) * inv_span : 0.0f;
    zinv[k] = zi;
    zmax = fmaxf(zmax, zi);
    fidx[k] = f;
  }
  zmax = fmaxf(zmax, EPSV_);                                  // clip(max, EPS)
  const float delta = fmaxf(__expf((EPSV_ - zmax) * inv_gamma), EPSV_);

  // ---------------- pass 2: weights + gathered feature blend ----------------
  float acc[DF];
#pragma unroll
  for (int c = 0; c < DF; ++c) acc[c] = 0.0f;
  float denom = delta;

#pragma unroll
  for (int k = 0; k < KF; ++k) {
    const float w = prob[k] * __expf((zinv[k] - zmax) * inv_gamma);
    denom += w;
    // w == 0 (masked, or exp underflow with gamma=1e-4) contributes exactly 0:
    // skip the 3x64B vertex gathers (L2-resident tables) in that case.
    if (fidx[k] >= 0 && w > 0.0f) {
      const int* fv = faces + 3 * fidx[k];
      const int v0 = fv[0], v1 = fv[1], v2 = fv[2];
      const float b0 = baryP[3 * k + 0];
      const float b1 = baryP[3 * k + 1];
      const float b2 = baryP[3 * k + 2];
      const float w0 = w * b0, w1 = w * b1, w2 = w * b2;
      const float4* r0 = (const float4*)(vert + (size_t)v0 * DF);
      const float4* r1 = (const float4*)(vert + (size_t)v1 * DF);
      const float4* r2 = (const float4*)(vert + (size_t)v2 * DF);
#pragma unroll
      for (int j = 0; j < 4; ++j) {
        const float4 a = r0[j], b = r1[j], c4 = r2[j];
        acc[4 * j + 0] += w0 * a.x + w1 * b.x + w2 * c4.x;
        acc[4 * j + 1] += w0 * a.y + w1 * b.y + w2 * c4.y;
        acc[4 * j + 2] += w0 * a.z + w1 * b.z + w2 * c4.z;
        acc[4 * j + 3] += w0 * a.w + w1 * b.w + w2 * c4.w;
      }
    }
  }

  // ---------------- output: (N, 17, H, W), coalesced across lanes ----------
  const float rden  = 1.0f / denom;
  const float alpha = 1.0f - omp;
  const int   n  = p >> 16;          // HW_ = 65536
  const int   hw = p & (HW_ - 1);
  float* op = out + (size_t)n * (NCH * (size_t)HW_) + hw;
#pragma unroll
  for (int c = 0; c < DF; ++c) op[(size_t)c * HW_] = acc[c] * rden;
  op[(size_t)DF * HW_] = alpha;
}

extern "C" void kernel_launch(void* const* d_in, const int* in_sizes, int n_in,
                              void* d_out, int out_size, void* d_ws, size_t ws_size,
                              hipStream_t stream) {
  (void)in_sizes; (void)n_in; (void)out_size; (void)d_ws; (void)ws_size;
  const float* vert  = (const float*)d_in[0];   // vert_features (V,16) f32
  const float* bary  = (const float*)d_in[1];   // bary_coords (N,H,W,K,3) f32
  const float* dists = (const float*)d_in[2];   // dists (N,H,W,K) f32
  const float* zbuf  = (const float*)d_in[3];   // zbuf  (N,H,W,K) f32
  const int*   faces = (const int*)d_in[4];     // faces (F,3) int
  const int*   ptf   = (const int*)d_in[5];     // pix_to_face (N,H,W,K) int
  float* out = (float*)d_out;                   // (N, 17, H, W) f32

  dim3 grid(NPIX / TPB);   // 1024 blocks, 256 threads = 8 wave32 each
  softblend_kernel<<<grid, TPB, 0, stream>>>(vert, bary, dists, zbuf, faces, ptf, out);
}